// ObjectPNEncoder_76587856822586
// MI455X (gfx1250) — compile-verified
//
#include <hip/hip_runtime.h>
#include <hip/hip_bf16.h>

typedef __attribute__((ext_vector_type(2))) float v2f;
typedef __attribute__((ext_vector_type(8))) float v8f;

#define BN_EPS 1e-5f

// ---------------------------------------------------------------------------
// Split (B,6,N) -> pts (B,N,3), feat (B,N,3)
// ---------------------------------------------------------------------------
__global__ void split_kernel(const float* __restrict__ xyz, float* __restrict__ pts,
                             float* __restrict__ feat, int B, int N) {
  long i = (long)blockIdx.x * blockDim.x + threadIdx.x;
  long total = (long)B * N * 3;
  if (i >= total) return;
  int d = (int)(i % 3);
  long bn = i / 3;
  int n = (int)(bn % N);
  int b = (int)(bn / N);
  pts[i]  = xyz[((long)b * 6 + d) * N + n];
  feat[i] = xyz[((long)b * 6 + 3 + d) * N + n];
}

// ---------------------------------------------------------------------------
// Farthest point sampling: one block per batch, dist[] lives in LDS
// ---------------------------------------------------------------------------
#define FPS_BLOCK 256
__global__ void fps_kernel(const float* __restrict__ pts, int N, int npoint,
                           int* __restrict__ idx_out) {
  int b = blockIdx.x;
  const float* P = pts + (long)b * N * 3;
  __shared__ float dist[4096];
  __shared__ float rv[FPS_BLOCK];
  __shared__ int   ri[FPS_BLOCK];
  __shared__ int   s_far;
  int t = threadIdx.x;
  for (int n = t; n < N; n += FPS_BLOCK) dist[n] = 1e10f;
  if (t == 0) s_far = 0;
  __syncthreads();
  for (int i = 0; i < npoint; ++i) {
    int far = s_far;
    if (t == 0) idx_out[b * npoint + i] = far;
    float cx = P[far * 3 + 0], cy = P[far * 3 + 1], cz = P[far * 3 + 2];
    float best = -1.0f; int bestn = N - 1;
    for (int n = t; n < N; n += FPS_BLOCK) {
      float dx = P[n * 3 + 0] - cx;
      float dy = P[n * 3 + 1] - cy;
      float dz = P[n * 3 + 2] - cz;
      float d = dx * dx + dy * dy + dz * dz;
      float m = fminf(dist[n], d);
      dist[n] = m;
      if (m > best) { best = m; bestn = n; }   // strict: keeps lowest n on ties
    }
    rv[t] = best; ri[t] = bestn;
    __syncthreads();
    for (int s = FPS_BLOCK / 2; s > 0; s >>= 1) {
      if (t < s) {
        if (rv[t + s] > rv[t] || (rv[t + s] == rv[t] && ri[t + s] < ri[t])) {
          rv[t] = rv[t + s]; ri[t] = ri[t + s];
        }
      }
      __syncthreads();
    }
    if (t == 0) s_far = ri[0];
    __syncthreads();
  }
}

// ---------------------------------------------------------------------------
// Gather sampled centers
// ---------------------------------------------------------------------------
__global__ void gather_xyz_kernel(const float* __restrict__ pts, const int* __restrict__ fidx,
                                  float* __restrict__ new_xyz, int B, int N, int S) {
  int i = blockIdx.x * blockDim.x + threadIdx.x;
  if (i >= B * S) return;
  int b = i / S;
  int n = fidx[i];
  const float* p = pts + ((long)b * N + n) * 3;
  new_xyz[i * 3 + 0] = p[0];
  new_xyz[i * 3 + 1] = p[1];
  new_xyz[i * 3 + 2] = p[2];
}

// ---------------------------------------------------------------------------
// Ball query: first K in-radius indices in index order (== sort semantics)
// ---------------------------------------------------------------------------
__global__ void ballquery_kernel(const float* __restrict__ pts, const float* __restrict__ new_xyz,
                                 int* __restrict__ idx, int B, int N, int S, int K, float r2) {
  int i = blockIdx.x * blockDim.x + threadIdx.x;
  if (i >= B * S) return;
  int b = i / S;
  const float* P = pts + (long)b * N * 3;
  float cx = new_xyz[i * 3 + 0], cy = new_xyz[i * 3 + 1], cz = new_xyz[i * 3 + 2];
  int* out = idx + (long)i * K;
  int cnt = 0;
  for (int n = 0; n < N && cnt < K; ++n) {
    float dx = P[n * 3 + 0] - cx;
    float dy = P[n * 3 + 1] - cy;
    float dz = P[n * 3 + 2] - cz;
    if (dx * dx + dy * dy + dz * dz <= r2) out[cnt++] = n;
  }
  int first = (cnt > 0) ? out[0] : 0;
  for (int k = cnt; k < K; ++k) out[k] = first;
}

// ---------------------------------------------------------------------------
// Build grouped rows: [pts[idx]-center , feat[idx]]  -> X (B*S*K, 3+Cf)
// ---------------------------------------------------------------------------
__global__ void group_build_kernel(const float* __restrict__ pts, const float* __restrict__ feat,
                                   int Cf, const float* __restrict__ new_xyz,
                                   const int* __restrict__ idx, float* __restrict__ X,
                                   int B, int N, int S, int K) {
  long i = (long)blockIdx.x * blockDim.x + threadIdx.x;
  long total = (long)B * S * K;
  if (i >= total) return;
  long bs = i / K;
  int b = (int)(bs / S);
  int n = idx[i];
  const float* p = pts + ((long)b * N + n) * 3;
  const float* c = new_xyz + bs * 3;
  int Cin = 3 + Cf;
  float* row = X + i * Cin;
  row[0] = p[0] - c[0];
  row[1] = p[1] - c[1];
  row[2] = p[2] - c[2];
  if (feat != nullptr) {
    const float* f = feat + ((long)b * N + n) * Cf;
    for (int ch = 0; ch < Cf; ++ch) row[3 + ch] = f[ch];
  }
}

// ---------------------------------------------------------------------------
// Concat rows for group-all:  X[row] = [A(row,0:Ca) , Bm(row,0:Cb)]
// ---------------------------------------------------------------------------
__global__ void concat_rows_kernel(const float* __restrict__ A, int Ca,
                                   const float* __restrict__ Bm, int Cb,
                                   float* __restrict__ X, long rows) {
  int Cin = Ca + Cb;
  long i = (long)blockIdx.x * blockDim.x + threadIdx.x;
  if (i >= rows * Cin) return;
  long r = i / Cin;
  int c = (int)(i % Cin);
  X[i] = (c < Ca) ? A[r * Ca + c] : Bm[r * Cb + (c - Ca)];
}

// ---------------------------------------------------------------------------
// GEMM via V_WMMA_F32_16X16X4_F32:  Y(R,C) = X(R,K) @ W(C,K)^T + bias
// 2x2 register-blocked: each wave owns a 32x32 output block (4 WMMA tiles),
// so one A-frag + one B-frag pair feeds two WMMAs each -> 2x the arithmetic
// intensity of naive tiling. All M/N-tile predicates are wave-uniform, so
// EXEC stays all-ones around every WMMA as the ISA requires.
// ---------------------------------------------------------------------------
__global__ void __launch_bounds__(256)
gemm_bias_wmma(const float* __restrict__ X, const float* __restrict__ W,
               const float* __restrict__ bias, float* __restrict__ Y,
               int R, int K, int C) {
  int wave = threadIdx.x >> 5;
  int lane = threadIdx.x & 31;
  long nBm = (R + 31) / 32;
  long nBn = C / 32;               // C is always a multiple of 32 here
  long blk = (long)blockIdx.x * 8 + wave;
  if (blk >= nBm * nBn) return;    // uniform within the wave
  int bm = (int)(blk / nBn);
  int bn = (int)(blk % nBn);
  int row0 = bm * 32;
  int col0 = bn * 32;
  bool m1 = (row0 + 16) < R;       // wave-uniform (R always a multiple of 16)
  int half = lane >> 4;            // selects K pair (ISA A/B layout)
  int l16 = lane & 15;
  const float* xrow0 = X + (long)(row0 + l16) * K;
  const float* xrow1 = X + (long)(row0 + 16 + l16) * K;   // only touched if m1
  const float* wrow0 = W + (long)(col0 + l16) * K;
  const float* wrow1 = W + (long)(col0 + 16 + l16) * K;
  v8f acc00 = {}, acc01 = {}, acc10 = {}, acc11 = {};
  int kfull = K & ~3;
  int k0 = 0;
  if (m1) {
    for (; k0 < kfull; k0 += 4) {
      int kk = k0 + 2 * half;
      v2f a0 = {xrow0[kk], xrow0[kk + 1]};
      v2f a1 = {xrow1[kk], xrow1[kk + 1]};
      v2f b0 = {wrow0[kk], wrow0[kk + 1]};
      v2f b1 = {wrow1[kk], wrow1[kk + 1]};
      acc00 = __builtin_amdgcn_wmma_f32_16x16x4_f32(false, a0, false, b0, (short)0, acc00, false, false);
      acc01 = __builtin_amdgcn_wmma_f32_16x16x4_f32(false, a0, false, b1, (short)0, acc01, false, false);
      acc10 = __builtin_amdgcn_wmma_f32_16x16x4_f32(false, a1, false, b0, (short)0, acc10, false, false);
      acc11 = __builtin_amdgcn_wmma_f32_16x16x4_f32(false, a1, false, b1, (short)0, acc11, false, false);
    }
    if (k0 < K) {                  // K tail (K in {6,131,259}) zero-padded
      int kk = k0 + 2 * half;
      v2f a0 = {0.f, 0.f}, a1 = {0.f, 0.f}, b0 = {0.f, 0.f}, b1 = {0.f, 0.f};
      if (kk < K)     { a0.x = xrow0[kk];     a1.x = xrow1[kk];
                        b0.x = wrow0[kk];     b1.x = wrow1[kk];     }
      if (kk + 1 < K) { a0.y = xrow0[kk + 1]; a1.y = xrow1[kk + 1];
                        b0.y = wrow0[kk + 1]; b1.y = wrow1[kk + 1]; }
      acc00 = __builtin_amdgcn_wmma_f32_16x16x4_f32(false, a0, false, b0, (short)0, acc00, false, false);
      acc01 = __builtin_amdgcn_wmma_f32_16x16x4_f32(false, a0, false, b1, (short)0, acc01, false, false);
      acc10 = __builtin_amdgcn_wmma_f32_16x16x4_f32(false, a1, false, b0, (short)0, acc10, false, false);
      acc11 = __builtin_amdgcn_wmma_f32_16x16x4_f32(false, a1, false, b1, (short)0, acc11, false, false);
    }
  } else {                         // single M-tile (FC head, R == 16)
    for (; k0 < kfull; k0 += 4) {
      int kk = k0 + 2 * half;
      v2f a0 = {xrow0[kk], xrow0[kk + 1]};
      v2f b0 = {wrow0[kk], wrow0[kk + 1]};
      v2f b1 = {wrow1[kk], wrow1[kk + 1]};
      acc00 = __builtin_amdgcn_wmma_f32_16x16x4_f32(false, a0, false, b0, (short)0, acc00, false, false);
      acc01 = __builtin_amdgcn_wmma_f32_16x16x4_f32(false, a0, false, b1, (short)0, acc01, false, false);
    }
    if (k0 < K) {
      int kk = k0 + 2 * half;
      v2f a0 = {0.f, 0.f}, b0 = {0.f, 0.f}, b1 = {0.f, 0.f};
      if (kk < K)     { a0.x = xrow0[kk];     b0.x = wrow0[kk];     b1.x = wrow1[kk];     }
      if (kk + 1 < K) { a0.y = xrow0[kk + 1]; b0.y = wrow0[kk + 1]; b1.y = wrow1[kk + 1]; }
      acc00 = __builtin_amdgcn_wmma_f32_16x16x4_f32(false, a0, false, b0, (short)0, acc00, false, false);
      acc01 = __builtin_amdgcn_wmma_f32_16x16x4_f32(false, a0, false, b1, (short)0, acc01, false, false);
    }
  }
  float bv0 = bias ? bias[col0 + l16] : 0.0f;
  float bv1 = bias ? bias[col0 + 16 + l16] : 0.0f;
  // D layout: VGPR v holds row base+v (lanes 0-15) / base+8+v (lanes 16-31)
#pragma unroll
  for (int v = 0; v < 8; ++v) {
    int r0 = row0 + v + half * 8;
    Y[(long)r0 * C + col0 + l16]      = acc00[v] + bv0;
    Y[(long)r0 * C + col0 + 16 + l16] = acc01[v] + bv1;
  }
  if (m1) {
#pragma unroll
    for (int v = 0; v < 8; ++v) {
      int r1 = row0 + 16 + v + half * 8;
      Y[(long)r1 * C + col0 + l16]      = acc10[v] + bv0;
      Y[(long)r1 * C + col0 + 16 + l16] = acc11[v] + bv1;
    }
  }
}

// ---------------------------------------------------------------------------
// Per-channel mean/var over all rows (BN reduce axes (0,1,2))
// ---------------------------------------------------------------------------
__global__ void bnstats_kernel(const float* __restrict__ Y, long R, int C,
                               float* __restrict__ mean, float* __restrict__ var) {
  int c = blockIdx.x;
  __shared__ float s1[256], s2[256];
  float a = 0.f, b = 0.f;
  for (long r = threadIdx.x; r < R; r += 256) {
    float v = Y[r * C + c];
    a += v; b += v * v;
  }
  s1[threadIdx.x] = a; s2[threadIdx.x] = b;
  __syncthreads();
  for (int s = 128; s > 0; s >>= 1) {
    if (threadIdx.x < s) { s1[threadIdx.x] += s1[threadIdx.x + s];
                           s2[threadIdx.x] += s2[threadIdx.x + s]; }
    __syncthreads();
  }
  if (threadIdx.x == 0) {
    float mu = s1[0] / (float)R;
    mean[c] = mu;
    var[c]  = s2[0] / (float)R - mu * mu;
  }
}

__global__ void bnrelu_kernel(float* __restrict__ Y, long R, int C,
                              const float* __restrict__ mean, const float* __restrict__ var,
                              const float* __restrict__ g, const float* __restrict__ be) {
  long i = (long)blockIdx.x * blockDim.x + threadIdx.x;
  if (i >= R * (long)C) return;
  int c = (int)(i % C);
  float x = (Y[i] - mean[c]) * rsqrtf(var[c] + BN_EPS) * g[c] + be[c];
  Y[i] = fmaxf(x, 0.0f);
}

// ---------------------------------------------------------------------------
// Max over the sample axis: (BS, K, C) -> (BS, C)
// ---------------------------------------------------------------------------
__global__ void maxpool_kernel(const float* __restrict__ Y, float* __restrict__ out,
                               long BS, int K, int C) {
  long i = (long)blockIdx.x * blockDim.x + threadIdx.x;
  if (i >= BS * (long)C) return;
  long bs = i / C;
  int c = (int)(i % C);
  const float* p = Y + (bs * K) * (long)C + c;
  float m = -3.402823466e38f;
  for (int k = 0; k < K; ++k) m = fmaxf(m, p[(long)k * C]);
  out[i] = m;
}

// ---------------------------------------------------------------------------
// Host-side helpers
// ---------------------------------------------------------------------------
static inline int cdiv(long a, long b) { return (int)((a + b - 1) / b); }

static void run_gemm(const float* X, const float* W, const float* bias, float* Y,
                     long R, int K, int C, hipStream_t s) {
  long tiles = ((R + 31) / 32) * (C / 32);
  gemm_bias_wmma<<<cdiv(tiles, 8), 256, 0, s>>>(X, W, bias, Y, (int)R, K, C);
}

static void run_layer(const float* X, int K, float* Y, int C, long R,
                      const float* W, const float* b, const float* g, const float* be,
                      float* mean, float* var, hipStream_t s) {
  run_gemm(X, W, b, Y, R, K, C, s);
  bnstats_kernel<<<C, 256, 0, s>>>(Y, R, C, mean, var);
  bnrelu_kernel<<<cdiv(R * (long)C, 256), 256, 0, s>>>(Y, R, C, mean, var, g, be);
}

extern "C" void kernel_launch(void* const* d_in, const int* in_sizes, int n_in,
                              void* d_out, int out_size, void* d_ws, size_t ws_size,
                              hipStream_t stream) {
  (void)in_sizes; (void)n_in; (void)out_size; (void)ws_size;
  const int B = 16, N = 4096;
  const int S1 = 512, K1 = 32;
  const int S2 = 128, K2 = 64;

  const float* xyz = (const float*)d_in[0];
  // sa1_params: d_in[1..12], sa2: [13..24], sa3: [25..36]
  auto PW  = [&](int base, int L) { return (const float*)d_in[base + 4 * L + 0]; };
  auto PB  = [&](int base, int L) { return (const float*)d_in[base + 4 * L + 1]; };
  auto PG  = [&](int base, int L) { return (const float*)d_in[base + 4 * L + 2]; };
  auto PBE = [&](int base, int L) { return (const float*)d_in[base + 4 * L + 3]; };
  const float* fc1_w = (const float*)d_in[37];
  const float* fc1_b = (const float*)d_in[38];
  const float* bn1_g = (const float*)d_in[39];
  const float* bn1_b = (const float*)d_in[40];
  const float* fc2_w = (const float*)d_in[41];
  const float* fc2_b = (const float*)d_in[42];
  float* out = (float*)d_out;

  // ---- workspace layout (float slots) ----
  float* ws = (float*)d_ws;
  long off = 0;
  float* pts    = ws + off; off += (long)B * N * 3;        // 196608
  float* feat   = ws + off; off += (long)B * N * 3;        // 196608
  float* l1_xyz = ws + off; off += (long)B * S1 * 3;       // 24576
  float* l1_p   = ws + off; off += (long)B * S1 * 128;     // 1048576
  float* l2_xyz = ws + off; off += (long)B * S2 * 3;       // 6144
  float* l2_p   = ws + off; off += (long)B * S2 * 256;     // 524288
  float* l3_p   = ws + off; off += (long)B * 1024;         // 16384
  float* fcbuf  = ws + off; off += (long)B * 512;          // 8192
  float* mean   = ws + off; off += 1024;
  float* var    = ws + off; off += 1024;
  int* fidx1 = (int*)(ws + off); off += (long)B * S1;            // 8192
  int* bidx1 = (int*)(ws + off); off += (long)B * S1 * K1;       // 262144
  int* fidx2 = (int*)(ws + off); off += (long)B * S2;            // 2048
  int* bidx2 = (int*)(ws + off); off += (long)B * S2 * K2;       // 131072
  float* bufA = ws + off; off += 33554432L;                 // max (262144,128)
  float* bufB = ws + off; off += 17170432L;                 // max (131072,131)

  // ---- split input ----
  split_kernel<<<cdiv((long)B * N * 3, 256), 256, 0, stream>>>(xyz, pts, feat, B, N);

  // =================== SA1 ===================
  long R1 = (long)B * S1 * K1;   // 262144
  fps_kernel<<<B, FPS_BLOCK, 0, stream>>>(pts, N, S1, fidx1);
  gather_xyz_kernel<<<cdiv((long)B * S1, 256), 256, 0, stream>>>(pts, fidx1, l1_xyz, B, N, S1);
  ballquery_kernel<<<cdiv((long)B * S1, 256), 256, 0, stream>>>(pts, l1_xyz, bidx1,
                                                                B, N, S1, K1, 0.2f * 0.2f);
  group_build_kernel<<<cdiv(R1, 256), 256, 0, stream>>>(pts, feat, 3, l1_xyz, bidx1,
                                                        bufB, B, N, S1, K1);
  run_layer(bufB, 6,  bufA, 64,  R1, PW(1,0), PB(1,0), PG(1,0), PBE(1,0), mean, var, stream);
  run_layer(bufA, 64, bufB, 64,  R1, PW(1,1), PB(1,1), PG(1,1), PBE(1,1), mean, var, stream);
  run_layer(bufB, 64, bufA, 128, R1, PW(1,2), PB(1,2), PG(1,2), PBE(1,2), mean, var, stream);
  maxpool_kernel<<<cdiv((long)B * S1 * 128, 256), 256, 0, stream>>>(bufA, l1_p,
                                                                    (long)B * S1, K1, 128);

  // =================== SA2 ===================
  long R2 = (long)B * S2 * K2;   // 131072
  fps_kernel<<<B, FPS_BLOCK, 0, stream>>>(l1_xyz, S1, S2, fidx2);
  gather_xyz_kernel<<<cdiv((long)B * S2, 256), 256, 0, stream>>>(l1_xyz, fidx2, l2_xyz, B, S1, S2);
  ballquery_kernel<<<cdiv((long)B * S2, 256), 256, 0, stream>>>(l1_xyz, l2_xyz, bidx2,
                                                                B, S1, S2, K2, 0.4f * 0.4f);
  group_build_kernel<<<cdiv(R2, 256), 256, 0, stream>>>(l1_xyz, l1_p, 128, l2_xyz, bidx2,
                                                        bufB, B, S1, S2, K2);
  run_layer(bufB, 131, bufA, 128, R2, PW(13,0), PB(13,0), PG(13,0), PBE(13,0), mean, var, stream);
  run_layer(bufA, 128, bufB, 128, R2, PW(13,1), PB(13,1), PG(13,1), PBE(13,1), mean, var, stream);
  run_layer(bufB, 128, bufA, 256, R2, PW(13,2), PB(13,2), PG(13,2), PBE(13,2), mean, var, stream);
  maxpool_kernel<<<cdiv((long)B * S2 * 256, 256), 256, 0, stream>>>(bufA, l2_p,
                                                                    (long)B * S2, K2, 256);

  // =================== SA3 (group all) ===================
  long R3 = (long)B * S2;        // 2048 rows of 259 channels
  concat_rows_kernel<<<cdiv(R3 * 259, 256), 256, 0, stream>>>(l2_xyz, 3, l2_p, 256, bufB, R3);
  run_layer(bufB, 259, bufA, 256,  R3, PW(25,0), PB(25,0), PG(25,0), PBE(25,0), mean, var, stream);
  run_layer(bufA, 256, bufB, 512,  R3, PW(25,1), PB(25,1), PG(25,1), PBE(25,1), mean, var, stream);
  run_layer(bufB, 512, bufA, 1024, R3, PW(25,2), PB(25,2), PG(25,2), PBE(25,2), mean, var, stream);
  maxpool_kernel<<<cdiv((long)B * 1024, 256), 256, 0, stream>>>(bufA, l3_p, (long)B, S2, 1024);

  // =================== FC head ===================
  run_gemm(l3_p, fc1_w, fc1_b, fcbuf, B, 1024, 512, stream);
  bnstats_kernel<<<512, 256, 0, stream>>>(fcbuf, B, 512, mean, var);
  bnrelu_kernel<<<cdiv((long)B * 512, 256), 256, 0, stream>>>(fcbuf, B, 512, mean, var,
                                                              bn1_g, bn1_b);
  run_gemm(fcbuf, fc2_w, fc2_b, out, B, 512, 512, stream);
}